// GraphTransformer_53291954208988
// MI455X (gfx1250) — compile-verified
//
#include <hip/hip_runtime.h>
#include <math.h>

// ---------------------------------------------------------------------------
// Problem constants (match reference)
// ---------------------------------------------------------------------------
static constexpr int NN    = 50000;
static constexpr int EE    = 800000;
static constexpr int INF_  = 64;     // input feature dim
static constexpr int EDIM  = 16;

typedef float v2f __attribute__((ext_vector_type(2)));
typedef float v8f __attribute__((ext_vector_type(8)));

// ---------------------------------------------------------------------------
// Utility: fill a buffer with a 32-bit pattern (zeros, -inf, ...)
// ---------------------------------------------------------------------------
__global__ void fill_u32(unsigned* __restrict__ p, long n, unsigned v) {
    long i = (long)blockIdx.x * blockDim.x + threadIdx.x;
    if (i < n) p[i] = v;
}

// float atomic-max via signed/unsigned int ordering trick (init to -inf bits)
__device__ inline void atomicMaxF(float* addr, float val) {
    if (val >= 0.0f)
        atomicMax((int*)addr, __float_as_int(val));
    else
        atomicMin((unsigned int*)addr, __float_as_uint(val));
}

// ---------------------------------------------------------------------------
// Dense GEMM with V_WMMA_F32_16X16X4_F32:  Y[n,fout] = X[n,fin] @ W + bias
// block = (32,4): 4 waves, each owns one 16-wide column tile; A tile in LDS.
// fp32 A fragment layout (ISA 7.12.2): lanes 0-15 hold K={0,1}, lanes 16-31
// hold K={2,3}; VGPR0 = even K of the pair, VGPR1 = odd. D: VGPR r holds row
// M = r + 8*(lane>=16), column N = lane&15.
// ---------------------------------------------------------------------------
__global__ __launch_bounds__(128)
void gemm_bias_wmma(const float* __restrict__ X, const float* __restrict__ W,
                    const float* __restrict__ bias, float* __restrict__ Y,
                    int nrows, int fin, int fout) {
    __shared__ float sx[16 * 128];          // A tile: 16 rows x fin (<=128)
    const int row0 = blockIdx.x * 16;
    if (row0 >= nrows) return;

    const int tid = threadIdx.y * 32 + threadIdx.x;
    for (int i = tid; i < 16 * fin; i += 128) {
        int r = i / fin, cc = i % fin;
        sx[i] = X[(long)(row0 + r) * fin + cc];
    }
    __syncthreads();

    const int lane = threadIdx.x;
    const int m    = lane & 15;
    const int koff = (lane >> 4) << 1;                       // 0 or 2
    const int col  = (blockIdx.y * 4 + threadIdx.y) * 16 + m;

    v8f c = {0.f, 0.f, 0.f, 0.f, 0.f, 0.f, 0.f, 0.f};
    for (int kk = 0; kk < fin; kk += 4) {
        v2f a, b;
        a.x = sx[m * fin + kk + koff];                       // A[m][kk+koff]
        a.y = sx[m * fin + kk + koff + 1];                   // A[m][kk+koff+1]
        b.x = W[(long)(kk + koff)     * fout + col];         // B[kk+koff][col]
        b.y = W[(long)(kk + koff + 1) * fout + col];
        c = __builtin_amdgcn_wmma_f32_16x16x4_f32(
                /*neg_a=*/false, a, /*neg_b=*/false, b,
                /*c_mod=*/(short)0, c, /*reuse_a=*/false, /*reuse_b=*/false);
    }

    const float bb = bias ? bias[col] : 0.0f;
#pragma unroll
    for (int r = 0; r < 8; ++r) {
        int mm = r + ((lane >> 4) << 3);                     // D-row mapping
        Y[(long)(row0 + mm) * fout + col] = c[r] + bb;
    }
}

// ---------------------------------------------------------------------------
// Edge phase kernels: one wave per edge. Edge projection e = edge_attr @ We
// is recomputed in registers (64 MACs/lane) instead of materializing 410 MB.
// ---------------------------------------------------------------------------
template <int FOUT, int C, int H>
__global__ __launch_bounds__(256)
void edge_alpha(const int* __restrict__ src, const int* __restrict__ dstv,
                const float* __restrict__ eattr, const float* __restrict__ We,
                const float* __restrict__ q, const float* __restrict__ k,
                float* __restrict__ alpha, float* __restrict__ amax, int nE) {
    const int eid = blockIdx.x * blockDim.y + threadIdx.y;
    if (eid >= nE) return;
    const int lane = threadIdx.x;
    const int s = src[eid], d = dstv[eid];
    const float* ea = eattr + (long)eid * EDIM;

    float acc[H];
#pragma unroll
    for (int h = 0; h < H; ++h) acc[h] = 0.f;

#pragma unroll
    for (int r = 0; r < FOUT / 32; ++r) {
        const int j = lane + 32 * r;
        float e = 0.f;
#pragma unroll
        for (int t = 0; t < EDIM; ++t) e = fmaf(ea[t], We[t * FOUT + j], e);
        const float p = q[(long)d * FOUT + j] * (k[(long)s * FOUT + j] + e);
        acc[(32 * r) / C] += p;       // C>=32 -> head index is per-r constant
    }
#pragma unroll
    for (int h = 0; h < H; ++h)
#pragma unroll
        for (int mlt = 16; mlt >= 1; mlt >>= 1)
            acc[h] += __shfl_xor(acc[h], mlt, 32);

    if (lane == 0) {
        const float scale = rsqrtf((float)C);
#pragma unroll
        for (int h = 0; h < H; ++h) {
            const float a = acc[h] * scale;
            alpha[(long)eid * H + h] = a;
            atomicMaxF(&amax[(long)d * H + h], a);
        }
    }
}

template <int H>
__global__ __launch_bounds__(256)
void edge_exp(const int* __restrict__ dstv, float* __restrict__ alpha,
              const float* __restrict__ amax, float* __restrict__ den, int nE) {
    const long i = (long)blockIdx.x * blockDim.x + threadIdx.x;
    if (i >= (long)nE * H) return;
    const int e = (int)(i / H), h = (int)(i % H);
    const int d = dstv[e];
    const float ex = __expf(alpha[i] - amax[(long)d * H + h]);
    alpha[i] = ex;                                   // reuse buffer for exp
    atomicAdd(&den[(long)d * H + h], ex);
}

template <int FOUT, int C, int H>
__global__ __launch_bounds__(256)
void edge_aggregate(const int* __restrict__ src, const int* __restrict__ dstv,
                    const float* __restrict__ eattr, const float* __restrict__ We,
                    const float* __restrict__ v, const float* __restrict__ exw,
                    const float* __restrict__ den, float* __restrict__ out,
                    int nE) {
    const int eid = blockIdx.x * blockDim.y + threadIdx.y;
    if (eid >= nE) return;
    const int lane = threadIdx.x;
    const int s = src[eid], d = dstv[eid];
    const float* ea = eattr + (long)eid * EDIM;
    __builtin_prefetch(eattr + (long)(eid + blockDim.y) * EDIM, 0, 1);

    float w[H];
#pragma unroll
    for (int h = 0; h < H; ++h)
        w[h] = exw[(long)eid * H + h] / den[(long)d * H + h];

#pragma unroll
    for (int r = 0; r < FOUT / 32; ++r) {
        const int j = lane + 32 * r;
        float e = 0.f;
#pragma unroll
        for (int t = 0; t < EDIM; ++t) e = fmaf(ea[t], We[t * FOUT + j], e);
        const float vj = v[(long)s * FOUT + j] + e;
        atomicAdd(&out[(long)d * FOUT + j], w[(32 * r) / C] * vj);
    }
}

// ---------------------------------------------------------------------------
// Beta-gated skip: g = sigmoid([o, xr, o-xr] @ Wb);  x = g*xr + (1-g)*o
// (non-concat layer has H==1 so head-mean is identity; o row stride == S)
// ---------------------------------------------------------------------------
template <int S>
__global__ __launch_bounds__(256)
void node_finalize(const float* __restrict__ out, const float* __restrict__ xr,
                   const float* __restrict__ Wb, float* __restrict__ xnew,
                   int n) {
    const int node = blockIdx.x * blockDim.y + threadIdx.y;
    if (node >= n) return;
    const int lane = threadIdx.x;
    constexpr int R = S / 32;

    float o[R], xx[R], partial = 0.f;
#pragma unroll
    for (int r = 0; r < R; ++r) {
        const int j = lane + 32 * r;
        o[r]  = out[(long)node * S + j];
        xx[r] = xr[(long)node * S + j];
        partial += Wb[j] * o[r] + Wb[S + j] * xx[r] + Wb[2 * S + j] * (o[r] - xx[r]);
    }
#pragma unroll
    for (int mlt = 16; mlt >= 1; mlt >>= 1)
        partial += __shfl_xor(partial, mlt, 32);

    const float g = 1.f / (1.f + __expf(-partial));
#pragma unroll
    for (int r = 0; r < R; ++r) {
        const int j = lane + 32 * r;
        xnew[(long)node * S + j] = g * xx[r] + (1.f - g) * o[r];
    }
}

// ---------------------------------------------------------------------------
// BatchNorm (biased var) + ReLU
// ---------------------------------------------------------------------------
__global__ __launch_bounds__(256)
void bn_stats(const float* __restrict__ x, float* __restrict__ sum,
              float* __restrict__ sumsq, int n, int cdim) {
    const int tid = blockIdx.x * blockDim.x + threadIdx.x;
    const int nth = gridDim.x * blockDim.x;
    const int ch  = tid % cdim;
    float s = 0.f, s2 = 0.f;
    for (int row = tid / cdim; row < n; row += nth / cdim) {
        const float v = x[(long)row * cdim + ch];
        s += v; s2 += v * v;
    }
    atomicAdd(&sum[ch], s);
    atomicAdd(&sumsq[ch], s2);
}

__global__ __launch_bounds__(256)
void bn_apply(float* __restrict__ x, const float* __restrict__ sum,
              const float* __restrict__ sumsq, const float* __restrict__ gamma,
              const float* __restrict__ beta, int n, int cdim) {
    const long i = (long)blockIdx.x * blockDim.x + threadIdx.x;
    if (i >= (long)n * cdim) return;
    const int ch = (int)(i % cdim);
    const float m   = sum[ch] / (float)n;
    const float var = sumsq[ch] / (float)n - m * m;
    float v = (x[i] - m) * rsqrtf(var + 1e-5f) * gamma[ch] + beta[ch];
    x[i] = v > 0.f ? v : 0.f;
}

// ---------------------------------------------------------------------------
// Host launcher
// ---------------------------------------------------------------------------
// Input flattening (jax pytree, sorted dict keys):
//   0: x [N,64]   1: edge_index [2,E]   2: edge_attr [E,16]
//   3: bns0.beta 4: bns0.gamma 5: bns1.beta 6: bns1.gamma
//   7+10*i + {0:Wb, 1:We, 2:Wk, 3:Wq, 4:Ws, 5:Wv, 6:bk, 7:bq, 8:bs, 9:bv}
extern "C" void kernel_launch(void* const* d_in, const int* in_sizes, int n_in,
                              void* d_out, int out_size, void* d_ws, size_t ws_size,
                              hipStream_t stream) {
    (void)in_sizes; (void)n_in; (void)out_size; (void)ws_size;

    const float* x0    = (const float*)d_in[0];
    const int*   ei    = (const int*)d_in[1];
    const float* eattr = (const float*)d_in[2];
    const int* src = ei;
    const int* dst = ei + EE;

    // workspace carve-up (floats)
    float* ws    = (float*)d_ws;
    float* qb    = ws;               ws += (long)NN * 128;
    float* kb    = ws;               ws += (long)NN * 128;
    float* vb    = ws;               ws += (long)NN * 128;
    float* xrb   = ws;               ws += (long)NN * 128;
    float* outb  = ws;               ws += (long)NN * 128;
    float* xbuf  = ws;               ws += (long)NN * 128;
    float* alpha = ws;               ws += (long)EE * 4;
    float* den   = ws;               ws += (long)NN * 4;
    float* amax  = ws;               ws += (long)NN * 4;
    float* bnsum = ws;               ws += 128;
    float* bnsq  = ws;               ws += 128;

    const dim3 gblk(32, 4);                  // gemm block: 4 waves
    const dim3 eblk(32, 8);                  // edge/node block: 8 waves
    const int  egrid = (EE + 7) / 8;
    const int  ngrid = (NN + 7) / 8;

    auto fills = [&](void* p, long n, unsigned v) {
        fill_u32<<<(int)((n + 255) / 256), 256, 0, stream>>>((unsigned*)p, n, v);
    };

    const int FINS[3]  = {INF_, 128, 128};
    const int FOUTS[3] = {128, 128, 64};
    const int HS[3]    = {4, 4, 1};

    for (int L = 0; L < 3; ++L) {
        const int fin = FINS[L], fout = FOUTS[L], H = HS[L];
        const float* Wb = (const float*)d_in[7 + 10 * L + 0];
        const float* We = (const float*)d_in[7 + 10 * L + 1];
        const float* Wk = (const float*)d_in[7 + 10 * L + 2];
        const float* Wq = (const float*)d_in[7 + 10 * L + 3];
        const float* Ws = (const float*)d_in[7 + 10 * L + 4];
        const float* Wv = (const float*)d_in[7 + 10 * L + 5];
        const float* bk = (const float*)d_in[7 + 10 * L + 6];
        const float* bq = (const float*)d_in[7 + 10 * L + 7];
        const float* bs = (const float*)d_in[7 + 10 * L + 8];
        const float* bv = (const float*)d_in[7 + 10 * L + 9];
        const float* Xin = (L == 0) ? x0 : xbuf;

        // per-layer init: out accumulator = 0, softmax den = 0, max = -inf
        fills(outb, (long)NN * fout, 0u);
        fills(den,  (long)NN * H,    0u);
        fills(amax, (long)NN * H,    0xFF800000u);

        // dense projections via fp32 WMMA
        const dim3 ggrid(NN / 16, fout / 64);
        gemm_bias_wmma<<<ggrid, gblk, 0, stream>>>(Xin, Wq, bq, qb,  NN, fin, fout);
        gemm_bias_wmma<<<ggrid, gblk, 0, stream>>>(Xin, Wk, bk, kb,  NN, fin, fout);
        gemm_bias_wmma<<<ggrid, gblk, 0, stream>>>(Xin, Wv, bv, vb,  NN, fin, fout);
        gemm_bias_wmma<<<ggrid, gblk, 0, stream>>>(Xin, Ws, bs, xrb, NN, fin, fout);

        // edge attention + segment softmax + scatter aggregate
        if (L < 2) {
            edge_alpha<128, 32, 4><<<egrid, eblk, 0, stream>>>(
                src, dst, eattr, We, qb, kb, alpha, amax, EE);
            edge_exp<4><<<(int)(((long)EE * 4 + 255) / 256), 256, 0, stream>>>(
                dst, alpha, amax, den, EE);
            edge_aggregate<128, 32, 4><<<egrid, eblk, 0, stream>>>(
                src, dst, eattr, We, vb, alpha, den, outb, EE);
            node_finalize<128><<<ngrid, eblk, 0, stream>>>(outb, xrb, Wb, xbuf, NN);

            // BatchNorm + ReLU (biased variance) on xbuf
            const float* bbeta  = (const float*)d_in[3 + 2 * L];
            const float* bgamma = (const float*)d_in[4 + 2 * L];
            fills(bnsum, 128, 0u);
            fills(bnsq,  128, 0u);
            bn_stats<<<64, 256, 0, stream>>>(xbuf, bnsum, bnsq, NN, 128);
            bn_apply<<<(int)(((long)NN * 128 + 255) / 256), 256, 0, stream>>>(
                xbuf, bnsum, bnsq, bgamma, bbeta, NN, 128);
        } else {
            edge_alpha<64, 64, 1><<<egrid, eblk, 0, stream>>>(
                src, dst, eattr, We, qb, kb, alpha, amax, EE);
            edge_exp<1><<<(int)(((long)EE + 255) / 256), 256, 0, stream>>>(
                dst, alpha, amax, den, EE);
            edge_aggregate<64, 64, 1><<<egrid, eblk, 0, stream>>>(
                src, dst, eattr, We, vb, alpha, den, outb, EE);
            node_finalize<64><<<ngrid, eblk, 0, stream>>>(
                outb, xrb, Wb, (float*)d_out, NN);
        }
    }
}